// EMM_GPU_14104672600676
// MI455X (gfx1250) — compile-verified
//
#include <hip/hip_runtime.h>
#include <hip/hip_bf16.h>

typedef __attribute__((ext_vector_type(2))) float v2f;
typedef __attribute__((ext_vector_type(8))) float v8f;
typedef __attribute__((ext_vector_type(4))) unsigned int v4u;
typedef __attribute__((ext_vector_type(8))) int v8i;
typedef __attribute__((ext_vector_type(4))) int v4i;

#define H 2048
#define MB 512

// d_out layout (floats)
#define O_READ   0
#define O_FRFT   2048
#define O_WRFT   2560
#define O_FWFT   15360
#define O_WWFT   15872
#define O_MT     28672

// workspace layout (floats)
#define W_GF 0
#define W_GW 1
#define W_ADD 2
#define W_ERASE 3
#define W_MWF 4      // 100
#define W_MWW 104    // 196 (14x14)
#define W_FOC 300    // 100
#define W_WIDE 400   // 36

__device__ __forceinline__ float waveReduce(float v) {
#pragma unroll
  for (int off = 16; off >= 1; off >>= 1) v += __shfl_xor(v, off, 32);
  return v;
}

__device__ __forceinline__ float ht01(float x) { return fminf(fmaxf(x, 0.f), 1.f); }

// ---- Tensor Data Mover: 1-D row (nelem f32, contiguous) global -> LDS -------
// D# per CDNA5 ISA 8.3/8.4: group0 = {count=1, lds_addr, global_addr, type=2},
// group1 = {data_size=4B, tensor_dim0=nelem, tensor_dim1=1, tile_dim0=nelem,
//           tile_dim1=1, dim0_stride=nelem}.  nelem must be < 65536.
__device__ __forceinline__ void tdm_load_1d(unsigned lds_off, const void* gp, unsigned nelem) {
  unsigned long long ga = (unsigned long long)gp;
  v4u g0 = { 1u,                                   // count=1, user mode
             lds_off,                              // lds_addr (bytes)
             (unsigned)ga,                         // global_addr[31:0]
             ((unsigned)(ga >> 32) & 0x01FFFFFFu) | (2u << 30) }; // [56:32] | type=2
  v8i g1 = { (int)(2u << 16),                      // data_size = 4B, no multicast
             (int)((nelem & 0xFFFFu) << 16),       // tensor_dim0[15:0]
             (int)(((nelem >> 16) & 0xFFFFu) | (1u << 16)), // dim0[31:16], tensor_dim1=1
             (int)((nelem & 0xFFFFu) << 16),       // tile_dim0
             1,                                    // tile_dim1 = 1
             (int)nelem,                           // tensor_dim0_stride[31:0]
             0, 0 };
  v4i z4 = { 0, 0, 0, 0 };
#if __has_include(<hip/amd_detail/amd_gfx1250_TDM.h>)
  v8i z8 = { 0, 0, 0, 0, 0, 0, 0, 0 };
  __builtin_amdgcn_tensor_load_to_lds(g0, g1, z4, z4, z8, 0);
#else
  __builtin_amdgcn_tensor_load_to_lds(g0, g1, z4, z4, 0);
#endif
}

// ---------------- Kernel 1: gates + mwf + mww (300 dot products vs h) --------
__global__ void k1_small(const float* __restrict__ h,
                         const float* __restrict__ w_fg, const float* __restrict__ b_fg,
                         const float* __restrict__ w_wg, const float* __restrict__ b_wg,
                         const float* __restrict__ w_ag, const float* __restrict__ b_ag,
                         const float* __restrict__ w_eg, const float* __restrict__ b_eg,
                         const float* __restrict__ w_hfc, const float* __restrict__ b_hfc,
                         const float* __restrict__ w_hwc, const float* __restrict__ b_hwc,
                         float* __restrict__ ws) {
  __shared__ float red[256];
  const int b = blockIdx.x, t = threadIdx.x;
  const float* w;
  if (b == 0) w = w_fg;
  else if (b == 1) w = w_wg;
  else if (b == 2) w = w_ag;
  else if (b == 3) w = w_eg;
  else if (b < 104) w = w_hfc + (b - 4) * H;
  else w = w_hwc + (b - 104) * H;

  const float4* w4 = (const float4*)w;
  const float4* h4 = (const float4*)h;
  float4 a0 = w4[t], a1 = w4[t + 256];
  float4 x0 = h4[t], x1 = h4[t + 256];
  float acc = a0.x * x0.x + a0.y * x0.y + a0.z * x0.z + a0.w * x0.w
            + a1.x * x1.x + a1.y * x1.y + a1.z * x1.z + a1.w * x1.w;
  red[t] = acc;
  __syncthreads();
#pragma unroll
  for (int s = 128; s > 0; s >>= 1) {
    if (t < s) red[t] += red[t + s];
    __syncthreads();
  }
  if (t == 0) {
    float d = red[0];
    if (b == 0) {
      float s = 1.f / (1.f + __expf(-(d + b_fg[0])));
      ws[W_GF] = ht01(1.2f * s - 0.1f);
    } else if (b == 1) {
      float s = 1.f / (1.f + __expf(-(d + b_wg[0])));
      ws[W_GW] = ht01(1.2f * s - 0.1f);
    } else if (b == 2) {
      ws[W_ADD] = ht01(d + b_ag[0]);
    } else if (b == 3) {
      ws[W_ERASE] = ht01(d + b_eg[0]);
    } else if (b < 104) {
      ws[W_MWF + (b - 4)] = fmaxf(d + b_hfc[b - 4], 0.f);
    } else {
      ws[W_MWW + (b - 104)] = fmaxf(d + b_hwc[b - 104], 0.f);
    }
  }
}

// ---------------- Kernel 2: big streaming GEMV (26624 rows x 2048) ----------
// HBM-bound (~220 MB of weights @ 0.5 FLOP/byte): one wave per row, 8 waves
// per block, h staged in LDS, coalesced b128 weight streams, wave32 reduce.
__global__ void k2_big_gemv(const float* __restrict__ h,
                            const float* __restrict__ w_hrf, const float* __restrict__ b_hrf,
                            const float* __restrict__ w_hrw, const float* __restrict__ b_hrw,
                            const float* __restrict__ w_hwf, const float* __restrict__ b_hwf,
                            const float* __restrict__ w_hww, const float* __restrict__ b_hww,
                            const float* __restrict__ frf, const float* __restrict__ wrf,
                            const float* __restrict__ fwf, const float* __restrict__ wwf,
                            const float* __restrict__ ws, float* __restrict__ out) {
  __shared__ float sh[H];
  const int tid = threadIdx.x;
  float4* sh4 = (float4*)sh;
  const float4* h4 = (const float4*)h;
  sh4[tid] = h4[tid];
  sh4[tid + 256] = h4[tid + 256];
  __syncthreads();

  const int wave = tid >> 5, lane = tid & 31;
  const int row = blockIdx.x * 8 + wave;   // 3328 blocks * 8 = 26624, exact

  const float gf = ws[W_GF], gw = ws[W_GW];
  const float* w; float bias, gate, base; int oidx;
  if (row < 512) {
    w = w_hrf + row * H; bias = b_hrf[row]; gate = gf; base = frf[row]; oidx = O_FRFT + row;
  } else if (row < 13312) {
    int r = row - 512;
    w = w_hrw + (size_t)r * H; bias = b_hrw[r]; gate = gw; base = wrf[r]; oidx = O_WRFT + r;
  } else if (row < 13824) {
    int r = row - 13312;
    w = w_hwf + r * H; bias = b_hwf[r]; gate = gf; base = fwf[r]; oidx = O_FWFT + r;
  } else {
    int r = row - 13824;
    w = w_hww + (size_t)r * H; bias = b_hww[r]; gate = gw; base = wwf[r]; oidx = O_WWFT + r;
  }

  const float4* w4 = (const float4*)w;
  float acc = 0.f;
#pragma unroll
  for (int i = 0; i < 16; ++i) {
    float4 a = w4[i * 32 + lane];
    float4 x = sh4[i * 32 + lane];
    acc += a.x * x.x + a.y * x.y + a.z * x.z + a.w * x.w;
  }
  acc = waveReduce(acc);
  if (lane == 0) {
    float v = ht01(acc + bias);
    out[oidx] = v * gate + base * (1.f - gate);
  }
}

// ---------------- Kernel 3a: focused (100) + wide (36) -----------------------
__global__ void k3a_readfeat(const float* __restrict__ m,
                             const float* __restrict__ out,  // frf_t / wrf_t slices
                             float* __restrict__ ws) {
  const int wv = blockIdx.x * 8 + (threadIdx.x >> 5);  // 17*8 = 136, exact
  const int lane = threadIdx.x & 31;
  if (wv < 100) {
    const int p = wv;
    float acc = 0.f;
    for (int c = lane; c < MB; c += 32) acc += m[c * 100 + p] * out[O_FRFT + c];
    acc = waveReduce(acc);
    if (lane == 0) ws[W_FOC + p] = fmaxf(acc, 0.f);
  } else {
    const int q = wv - 100;
    const int qy = q / 6, qx = q % 6;
    float acc = 0.f;
    for (int c = lane; c < MB; c += 32) {
      const float* wr = out + O_WRFT + c * 25;
      const float* mc = m + c * 100;
#pragma unroll
      for (int ky = 0; ky < 5; ++ky)
#pragma unroll
        for (int kx = 0; kx < 5; ++kx)
          acc += mc[(qy + ky) * 10 + (qx + kx)] * wr[ky * 5 + kx];
    }
    acc = waveReduce(acc);
    if (lane == 0) ws[W_WIDE + q] = fmaxf(acc, 0.f);
  }
}

// ---------------- Kernel 3b: read = w_cwr@wide + b + relu(w_cfr@focused + b) -
__global__ void k3b_read(const float* __restrict__ w_cfr, const float* __restrict__ b_cfr,
                         const float* __restrict__ w_cwr, const float* __restrict__ b_cwr,
                         const float* __restrict__ ws, float* __restrict__ out) {
  const int r = blockIdx.x * 8 + (threadIdx.x >> 5);  // 256*8 = 2048, exact
  const int lane = threadIdx.x & 31;
  float af = 0.f, aw = 0.f;
  for (int p = lane; p < 100; p += 32) af += w_cfr[r * 100 + p] * ws[W_FOC + p];
  for (int p = lane; p < 36; p += 32)  aw += w_cwr[r * 36 + p] * ws[W_WIDE + p];
  af = waveReduce(af);
  aw = waveReduce(aw);
  if (lane == 0)
    out[O_READ + r] = (aw + b_cwr[r]) + fmaxf(af + b_cfr[r], 0.f);
}

// ---------------- Kernel 4: write path via TDM + WMMA f32 16x16x4 ------------
// wide_write = (512x25 wwf_t) x (25x100 mww patches): M=512, K=25(pad28),
// N=100(pad112).  TDM stages the A tile + mww into LDS; the wave builds a
// zero-padded A [16][28] and a transposed patch matrix B [16][28] once, so the
// inner loop is 2x ds_load_b64 + v_wmma per K chunk.  Fused with the
// focused-write outer product, erase/add blend and m_t = m + mem_write.
// One full wave32 per 16x16 tile => EXEC all-1s at every WMMA.
__global__ void k4_write_wmma(const float* __restrict__ m,
                              const float* __restrict__ ws,
                              float* __restrict__ out) {
  __shared__ __align__(16) float sRawA[16 * 25];   // TDM landing: wwf_t tile
  __shared__ __align__(16) float sMww[196];        // TDM landing: mww 14x14
  __shared__ __align__(16) float sA[16 * 28];      // A  [M=16][K=28], K-padded
  __shared__ __align__(16) float sB[16 * 28];      // B^T[N=16][K=28], K-padded
  const int lane = threadIdx.x;                    // block == one wave32
  const int half = lane >> 4, l16 = lane & 15;
  const int c0 = blockIdx.x * 16;                  // 32 M-tiles
  const int n0 = blockIdx.y * 16;                  // 7 N-tiles (last partial)

  const float* wwf_t = out + O_WWFT;               // [512][25]
  const float* fwf_t = out + O_FWFT;               // [512]
  const float* mwfp  = ws + W_MWF;                 // [100]
  const float erase  = ws[W_ERASE];
  const float addg   = ws[W_ADD];

  // --- TDM: async DMA of both operand blocks into LDS -----------------------
  tdm_load_1d((unsigned)(size_t)&sRawA[0], wwf_t + c0 * 25, 16 * 25);
  tdm_load_1d((unsigned)(size_t)&sMww[0], ws + W_MWW, 196);

  // zero K-pad of A while the DMA is in flight
  for (int i = lane; i < 16 * 28; i += 32) sA[i] = 0.f;

  __builtin_amdgcn_s_wait_tensorcnt(0);
  __syncthreads();

  // A: repack [16][25] -> [16][28] (zero tail)
  for (int i = lane; i < 16 * 25; i += 32) {
    int r = i / 25, k = i - r * 25;
    sA[r * 28 + k] = sRawA[i];
  }
  // B^T: sB[nn*28 + k] = mww[(oy+k/5)*14 + ox+k%5], 0 outside K<25 / N<100
  for (int i = lane; i < 16 * 28; i += 32) {
    int nn = i / 28, k = i - nn * 28;
    int n = n0 + nn;
    float v = 0.f;
    if (k < 25 && n < 100) {
      int oy = n / 10, ox = n - oy * 10;
      int ky = k / 5,  kx = k - ky * 5;
      v = sMww[(oy + ky) * 14 + (ox + kx)];
    }
    sB[i] = v;
  }
  __syncthreads();

  // --- WMMA main loop: A lane holds A[M=l16][K=kb..kb+1], B holds B[K][N=l16]
  v8f acc = {};
  const int kbase = 2 * half;
#pragma unroll
  for (int k0 = 0; k0 < 28; k0 += 4) {             // K = 25 padded to 28
    const int k = k0 + kbase;                      // even => 8B-aligned b64
    v2f Av = *(const v2f*)&sA[l16 * 28 + k];
    v2f Bv = *(const v2f*)&sB[l16 * 28 + k];
    acc = __builtin_amdgcn_wmma_f32_16x16x4_f32(
        /*neg_a=*/false, Av, /*neg_b=*/false, Bv,
        /*c_mod=*/(short)0, acc, /*reuse_a=*/false, /*reuse_b=*/false);
  }

  // --- epilogue: D[M=i+8*half][N=l16] ---------------------------------------
  const int n = n0 + l16;
  if (n < 100) {
    const float scale = 1.f - erase;
    const float mwfn = mwfp[n];
    const int cb = c0 + 8 * half;                  // 8 consecutive channels
    float4 f0 = *(const float4*)&fwf_t[cb];
    float4 f1 = *(const float4*)&fwf_t[cb + 4];
    float fw[8] = { f0.x, f0.y, f0.z, f0.w, f1.x, f1.y, f1.z, f1.w };
#pragma unroll
    for (int i = 0; i < 8; ++i) {
      const int c = cb + i;
      const float mem_write = (mwfn * fw[i] + acc[i]) * scale + 2.f * addg;
      out[O_MT + c * 100 + n] = m[c * 100 + n] + mem_write;
    }
  }
}

extern "C" void kernel_launch(void* const* d_in, const int* in_sizes, int n_in,
                              void* d_out, int out_size, void* d_ws, size_t ws_size,
                              hipStream_t stream) {
  const float* h     = (const float*)d_in[0];
  const float* frf   = (const float*)d_in[1];
  const float* wrf   = (const float*)d_in[2];
  const float* fwf   = (const float*)d_in[3];
  const float* wwf   = (const float*)d_in[4];
  const float* m     = (const float*)d_in[5];
  const float* w_fg  = (const float*)d_in[6];
  const float* b_fg  = (const float*)d_in[7];
  const float* w_wg  = (const float*)d_in[8];
  const float* b_wg  = (const float*)d_in[9];
  const float* w_hrf = (const float*)d_in[10];
  const float* b_hrf = (const float*)d_in[11];
  const float* w_hrw = (const float*)d_in[12];
  const float* b_hrw = (const float*)d_in[13];
  const float* w_cfr = (const float*)d_in[14];
  const float* b_cfr = (const float*)d_in[15];
  const float* w_cwr = (const float*)d_in[16];
  const float* b_cwr = (const float*)d_in[17];
  const float* w_hfc = (const float*)d_in[18];
  const float* b_hfc = (const float*)d_in[19];
  const float* w_hwc = (const float*)d_in[20];
  const float* b_hwc = (const float*)d_in[21];
  const float* w_ag  = (const float*)d_in[22];
  const float* b_ag  = (const float*)d_in[23];
  const float* w_eg  = (const float*)d_in[24];
  const float* b_eg  = (const float*)d_in[25];
  const float* w_hwf = (const float*)d_in[26];
  const float* b_hwf = (const float*)d_in[27];
  const float* w_hww = (const float*)d_in[28];
  const float* b_hww = (const float*)d_in[29];

  float* out = (float*)d_out;
  float* ws  = (float*)d_ws;

  k1_small<<<300, 256, 0, stream>>>(h, w_fg, b_fg, w_wg, b_wg, w_ag, b_ag, w_eg, b_eg,
                                    w_hfc, b_hfc, w_hwc, b_hwc, ws);
  k2_big_gemv<<<3328, 256, 0, stream>>>(h, w_hrf, b_hrf, w_hrw, b_hrw,
                                        w_hwf, b_hwf, w_hww, b_hww,
                                        frf, wrf, fwf, wwf, ws, out);
  k3a_readfeat<<<17, 256, 0, stream>>>(m, out, ws);
  k3b_read<<<256, 256, 0, stream>>>(w_cfr, b_cfr, w_cwr, b_cwr, ws, out);
  k4_write_wmma<<<dim3(32, 7), 32, 0, stream>>>(m, ws, out);
}